// GNNML3Model_15238543966414
// MI455X (gfx1250) — compile-verified
//
#include <hip/hip_runtime.h>

typedef __attribute__((ext_vector_type(2))) float v2f;
typedef __attribute__((ext_vector_type(8))) float v8f;

constexpr int Nn   = 50000;
constexpr int Ee   = 800000;
constexpr int S    = 5;
constexpr int CIN  = 128;
constexpr int CC   = 64;
constexpr int CM   = 64;
constexpr int L    = 3;
constexpr int COUT = 64;
constexpr int NCOL = S * CC + CM + CM;   // 448: [xW | h@W5 | h@W6]
constexpr int LDS_STRIDE = CIN + 4;      // 132: transposed B slice, conflict-free

// ---------------------------------------------------------------------------
// Edge gating MLP: C_tilde[e,:] = relu(concat(sig(cp@W1+b1), sig(cp@W2+b2)*sig(cp@W3+b3)) @ W4 + b4)
// ---------------------------------------------------------------------------
__global__ void edge_gate_kernel(const float* __restrict__ Cp,
                                 const float* __restrict__ W1, const float* __restrict__ b1,
                                 const float* __restrict__ W2, const float* __restrict__ b2,
                                 const float* __restrict__ W3, const float* __restrict__ b3,
                                 const float* __restrict__ W4, const float* __restrict__ b4,
                                 float* __restrict__ Ct, int E)
{
    int e = blockIdx.x * blockDim.x + threadIdx.x;
    if (e >= E) return;
    float cp[S];
#pragma unroll
    for (int i = 0; i < S; ++i) cp[i] = Cp[e * S + i];

    float o1[S], o23[S];
#pragma unroll
    for (int j = 0; j < S; ++j) {
        float a1 = b1[j], a2 = b2[j], a3 = b3[j];
#pragma unroll
        for (int i = 0; i < S; ++i) {
            a1 += cp[i] * W1[i * S + j];
            a2 += cp[i] * W2[i * S + j];
            a3 += cp[i] * W3[i * S + j];
        }
        float s1 = 1.0f / (1.0f + __expf(-a1));
        float s2 = 1.0f / (1.0f + __expf(-a2));
        float s3 = 1.0f / (1.0f + __expf(-a3));
        o1[j]  = s1;
        o23[j] = s2 * s3;
    }
#pragma unroll
    for (int j = 0; j < S; ++j) {
        float a = b4[j];
#pragma unroll
        for (int i = 0; i < S; ++i) {
            a += o1[i]  * W4[i * S + j];
            a += o23[i] * W4[(S + i) * S + j];
        }
        Ct[e * S + j] = fmaxf(a, 0.0f);
    }
}

// ---------------------------------------------------------------------------
// Repack per-layer weights into one row-major B [CIN x NCOL]:
//   cols [0,320): B[c][s*64+o] = Wc[s][c][o];  [320,384): W5;  [384,448): W6
// ---------------------------------------------------------------------------
__global__ void pack_B_kernel(const float* __restrict__ Wc,
                              const float* __restrict__ W5,
                              const float* __restrict__ W6,
                              float* __restrict__ B)
{
    int idx = blockIdx.x * blockDim.x + threadIdx.x;
    if (idx >= CIN * NCOL) return;
    int c   = idx / NCOL;
    int col = idx % NCOL;
    float v;
    if (col < S * CC) {
        int s = col >> 6, o = col & 63;
        v = Wc[((size_t)s * CIN + c) * CC + o];
    } else if (col < S * CC + CM) {
        v = W5[(size_t)c * CM + (col - S * CC)];
    } else {
        v = W6[(size_t)c * CM + (col - S * CC - CM)];
    }
    B[idx] = v;
}

// ---------------------------------------------------------------------------
// fp32 WMMA GEMM: D[M x NC] = A[M x 128] * B[128 x NC] (+ bias), NC compile-time.
// One wave per 16x16 tile (4 waves/block cover a 16x64 slab of D).
// B slice for the block's 64 columns is staged TRANSPOSED in LDS
// (ldsB[col][k], stride 132) so each K-step's B fragment is one ds_load_b64,
// conflict-free across both lane halves (bank = (4*col + k) mod 64).
// Grids are exact -> EXEC all-ones around V_WMMA (ISA requirement).
// ---------------------------------------------------------------------------
template <int NC>
__global__ __launch_bounds__(128) void gemm_wmma_kernel(
    const float* __restrict__ A, const float* __restrict__ B,
    float* __restrict__ D, const float* __restrict__ bias)
{
    __shared__ float ldsB[64 * LDS_STRIDE];       // 33 KB

    const int tid    = threadIdx.x;
    const int lane   = tid & 31;
    const int wave   = tid >> 5;
    const int mBase  = blockIdx.x * 16;
    const int nBase0 = blockIdx.y * 64;

    // Cooperative transposed fill: element idx = it*128 + tid; coalesced global
    // reads (64 consecutive cols), conflict-free ds stores (bank = 4c + r).
    {
        const int c     = tid & 63;
        const int rBase = tid >> 6;               // 0 or 1
        const float* g  = B + (size_t)rBase * NC + nBase0 + c;
        float* l        = ldsB + c * LDS_STRIDE + rBase;
#pragma unroll 16
        for (int it = 0; it < 64; ++it)
            l[it * 2] = g[(size_t)(it * 2) * NC];
    }
    __syncthreads();

    const int col = (wave << 4) + (lane & 15);    // block-local column 0..63
    const int kh  = (lane >> 4) << 1;             // 0 or 2 (K half select)

    const float* aRow = A + (size_t)(mBase + (lane & 15)) * CIN + kh;
    const float* bLds = ldsB + col * LDS_STRIDE + kh;

    v8f c = {};
#pragma unroll
    for (int k = 0; k < CIN; k += 4) {
        v2f a, b;
        a.x = aRow[k];
        a.y = aRow[k + 1];
        b = *(const v2f*)&bLds[k];                // ds_load_b64: B[k+kh][col], B[k+kh+1][col]
        c = __builtin_amdgcn_wmma_f32_16x16x4_f32(
                /*neg_a=*/false, a, /*neg_b=*/false, b,
                /*c_mod=*/(short)0, c, /*reuse_a=*/false, /*reuse_b=*/false);
    }

    const int gcol = nBase0 + col;
    const float bv = bias ? bias[gcol] : 0.0f;
    float* dPtr = D + (size_t)(mBase + ((lane >> 4) << 3)) * NC + gcol;
#pragma unroll
    for (int r = 0; r < 8; ++r)
        dPtr[(size_t)r * NC] = c[r] + bv;         // immediate-offset stores
}

// ---------------------------------------------------------------------------
__global__ void zero_kernel(float* __restrict__ p, int n)
{
    int i = blockIdx.x * blockDim.x + threadIdx.x;
    if (i < n) p[i] = 0.0f;
}

// ---------------------------------------------------------------------------
// Weighted gather + scatter-add, 16 threads/edge, float4 per thread:
//   msg[e, 4q:4q+4] = sum_s Ct[e,s] * xW[src[e]][s*64 + 4q : 4q+4]
// ---------------------------------------------------------------------------
__global__ void edge_scatter_kernel(const float* __restrict__ Ct,
                                    const float* __restrict__ raw,   // [N, NCOL], cols 0..319 = xW
                                    const int* __restrict__ src,
                                    const int* __restrict__ dst,
                                    float* __restrict__ aggr, int E)
{
    long long t = (long long)blockIdx.x * blockDim.x + threadIdx.x;
    int e = (int)(t >> 4);
    int q = (int)(t & 15);
    if (e >= E) return;

    float ct[S];
#pragma unroll
    for (int s = 0; s < S; ++s) ct[s] = Ct[(size_t)e * S + s];

    const float4* xw = (const float4*)(raw + (size_t)src[e] * NCOL) + q;
    float4 acc = make_float4(0.f, 0.f, 0.f, 0.f);
#pragma unroll
    for (int s = 0; s < S; ++s) {
        float4 v = xw[s * 16];                    // global_load_b128, coalesced
        acc.x += ct[s] * v.x;
        acc.y += ct[s] * v.y;
        acc.z += ct[s] * v.z;
        acc.w += ct[s] * v.w;
    }
    float* ap = aggr + (size_t)dst[e] * CC + q * 4;
    atomicAdd(ap + 0, acc.x);
    atomicAdd(ap + 1, acc.y);
    atomicAdd(ap + 2, acc.z);
    atomicAdd(ap + 3, acc.w);
}

// ---------------------------------------------------------------------------
// h_next[n, 0:64] = relu(aggr);  h_next[n, 64:128] = relu((m5+b5)*(m6+b6))
// float4 per thread (N*32 groups).
// ---------------------------------------------------------------------------
__global__ void combine_kernel(const float* __restrict__ aggr,
                               const float* __restrict__ raw,
                               const float* __restrict__ b5,
                               const float* __restrict__ b6,
                               float* __restrict__ hNext, int N)
{
    int t = blockIdx.x * blockDim.x + threadIdx.x;
    if (t >= N * 32) return;
    int n = t >> 5, g = t & 31;
    float4 v;
    if (g < 16) {
        v = ((const float4*)(aggr + (size_t)n * CC))[g];
    } else {
        int o = g - 16;
        float4 m5 = ((const float4*)(raw + (size_t)n * NCOL + S * CC))[o];
        float4 m6 = ((const float4*)(raw + (size_t)n * NCOL + S * CC + CM))[o];
        float4 a5 = ((const float4*)b5)[o];
        float4 a6 = ((const float4*)b6)[o];
        v.x = (m5.x + a5.x) * (m6.x + a6.x);
        v.y = (m5.y + a5.y) * (m6.y + a6.y);
        v.z = (m5.z + a5.z) * (m6.z + a6.z);
        v.w = (m5.w + a5.w) * (m6.w + a6.w);
    }
    v.x = fmaxf(v.x, 0.f);
    v.y = fmaxf(v.y, 0.f);
    v.z = fmaxf(v.z, 0.f);
    v.w = fmaxf(v.w, 0.f);
    ((float4*)hNext)[t] = v;
}

// ---------------------------------------------------------------------------
extern "C" void kernel_launch(void* const* d_in, const int* in_sizes, int n_in,
                              void* d_out, int out_size, void* d_ws, size_t ws_size,
                              hipStream_t stream)
{
    const float* x   = (const float*)d_in[0];
    const float* Cp  = (const float*)d_in[1];
    const int*   ei  = (const int*)  d_in[2];
    const float* W1  = (const float*)d_in[3];
    const float* b1  = (const float*)d_in[4];
    const float* W2  = (const float*)d_in[5];
    const float* b2  = (const float*)d_in[6];
    const float* W3  = (const float*)d_in[7];
    const float* b3  = (const float*)d_in[8];
    const float* W4  = (const float*)d_in[9];
    const float* b4  = (const float*)d_in[10];
    const float* Wc  = (const float*)d_in[11];
    const float* W5  = (const float*)d_in[12];
    const float* b5  = (const float*)d_in[13];
    const float* W6  = (const float*)d_in[14];
    const float* b6  = (const float*)d_in[15];
    const float* Wr  = (const float*)d_in[16];
    const float* br  = (const float*)d_in[17];
    (void)in_sizes; (void)n_in; (void)out_size; (void)ws_size;

    const int* src = ei;
    const int* dst = ei + Ee;

    // workspace layout (all offsets 16B-aligned)
    char* ws = (char*)d_ws;
    float* Ct   = (float*)ws; ws += (size_t)Ee * S * sizeof(float);       // 16.0 MB
    float* Bp   = (float*)ws; ws += (size_t)CIN * NCOL * sizeof(float);   // 0.23 MB
    float* raw  = (float*)ws; ws += (size_t)Nn * NCOL * sizeof(float);    // 89.6 MB
    float* aggr = (float*)ws; ws += (size_t)Nn * CC * sizeof(float);      // 12.8 MB
    float* h0   = (float*)ws; ws += (size_t)Nn * CIN * sizeof(float);     // 25.6 MB
    float* h1   = (float*)ws;                                             // 25.6 MB
    float* hBuf[2] = {h0, h1};

    const float* hIn = x;
    for (int l = 0; l < L; ++l) {
        edge_gate_kernel<<<(Ee + 255) / 256, 256, 0, stream>>>(
            Cp, W1 + l * S * S, b1 + l * S, W2 + l * S * S, b2 + l * S,
            W3 + l * S * S, b3 + l * S, W4 + l * 2 * S * S, b4 + l * S, Ct, Ee);

        pack_B_kernel<<<(CIN * NCOL + 255) / 256, 256, 0, stream>>>(
            Wc + (size_t)l * S * CIN * CC,
            W5 + (size_t)l * CIN * CM,
            W6 + (size_t)l * CIN * CM, Bp);

        dim3 g(Nn / 16, NCOL / 64);               // 3125 x 7, 4 waves/block
        gemm_wmma_kernel<NCOL><<<g, 128, 0, stream>>>(hIn, Bp, raw, nullptr);

        zero_kernel<<<(Nn * CC + 255) / 256, 256, 0, stream>>>(aggr, Nn * CC);

        long long tot = (long long)Ee * 16;
        edge_scatter_kernel<<<(unsigned)((tot + 255) / 256), 256, 0, stream>>>(
            Ct, raw, src, dst, aggr, Ee);

        combine_kernel<<<(Nn * 32 + 255) / 256, 256, 0, stream>>>(
            aggr, raw, b5 + l * CM, b6 + l * CM, hBuf[l & 1], Nn);

        hIn = hBuf[l & 1];
    }

    // readout: out = h @ Wr + br   (Wr already [128,64] row-major)
    dim3 g2(Nn / 16, COUT / 64);
    gemm_wmma_kernel<COUT><<<g2, 128, 0, stream>>>(hIn, Wr, (float*)d_out, br);
}